// ScaledDotProductAttention_42271068127847
// MI455X (gfx1250) — compile-verified
//
#include <hip/hip_runtime.h>
#include <hip/hip_bf16.h>

// Scaled dot-product attention (fp32), B=2 H=16 S=2048 D=64, with mask,
// softmax, dropout(0.1, inverted scaling), returning (out, p_attn).
// Strategy: one 16-row q-stripe per 256-thread workgroup; the full
// 16x2048 score stripe lives in LDS (128KB of the WGP's 320KB), so raw
// scores never touch HBM. Both matmuls use V_WMMA_F32_16X16X4_F32.

#define B_ 2
#define H_ 16
#define S_ 2048
#define D_ 64
#define QT 16                      // q rows per workgroup
#define NEGV -1000000000.0f
#define SCALE 0.125f               // 1/sqrt(64)
#define INV_KEEP (1.0f / 0.9f)
#define KEEP_THRESH 3865470566u    // floor(0.9 * 2^32)

typedef float v2f __attribute__((ext_vector_type(2)));
typedef float v8f __attribute__((ext_vector_type(8)));

__device__ __forceinline__ v8f wmma4(v2f a, v2f b, v8f c) {
  // D = A(16x4,f32) * B(4x16,f32) + C(16x16,f32); 8-arg VOP3P form.
  return __builtin_amdgcn_wmma_f32_16x16x4_f32(
      /*neg_a=*/false, a, /*neg_b=*/false, b,
      /*c_mod=*/(short)0, c, /*reuse_a=*/false, /*reuse_b=*/false);
}

__device__ __forceinline__ float dropf(float p, unsigned idx) {
  // PCG-style hash -> uniform u32; deterministic per flat p_attn index.
  unsigned s = idx * 747796405u + 2891336453u;
  unsigned w = ((s >> ((s >> 28) + 4u)) ^ s) * 277803737u;
  w = (w >> 22) ^ w;
  return (w < KEEP_THRESH) ? p * INV_KEEP : 0.0f;
}

__global__ void __launch_bounds__(256, 1)
sdpa_f32_wmma_kernel(const float* __restrict__ Qg,
                     const float* __restrict__ Kg,
                     const float* __restrict__ Vg,
                     const int*   __restrict__ Mg,
                     float* __restrict__ outg,
                     float* __restrict__ pattng) {
  extern __shared__ float smem[];            // [QT*S_] scores + [S_] mask/red
  int*   mask_s = (int*)(smem + QT * S_);
  float* red    = smem + QT * S_;            // reused in phase 3

  const int blk = blockIdx.x;                // B_*H_*(S_/QT) = 4096 blocks
  const int qt  = blk & (S_ / QT - 1);       // 0..127
  const int bh  = blk >> 7;                  // 0..31
  const int b   = bh >> 4;                   // /H_
  const int q0  = qt * QT;

  const float* Q = Qg + (size_t)bh * S_ * D_;
  const float* K = Kg + (size_t)bh * S_ * D_;
  const float* V = Vg + (size_t)bh * S_ * D_;
  float* outp  = outg   + ((size_t)bh * S_ + q0) * D_;
  float* pattn = pattng + ((size_t)bh * S_ + q0) * S_;

  const int tid  = threadIdx.x;
  const int w    = tid >> 5;                 // wave 0..7
  const int lane = tid & 31;
  const int m    = lane & 15;
  const int hi   = lane >> 4;

  // ---- stage mask row [S_] into LDS ----
  const int* mrow = Mg + (size_t)b * S_;
  for (int i = tid; i < S_; i += 256) mask_s[i] = mrow[i];
  __syncthreads();

  // ================= Phase 1: scores = (Q K^T)*scale, mask -> LDS ========
  {
    // Q fragment (A matrix), reused across all 16 k-tiles of this wave:
    // lane m holds row q0+m; VGPR pair s covers d = 4s + 2*hi + {0,1}.
    const float* qrow = Q + (size_t)(q0 + m) * D_;
    v2f qa[16];
#pragma unroll
    for (int s = 0; s < 16; ++s)
      qa[s] = *(const v2f*)(qrow + 4 * s + 2 * hi);

    for (int j = 0; j < 16; ++j) {           // this wave's 16 k-tiles
      const int k0 = (w + 8 * j) * 16;
      const float* krow = K + (size_t)(k0 + m) * D_;
      v8f acc = {};
#pragma unroll
      for (int s = 0; s < 16; ++s) {         // reduce over D in K=4 chunks
        const v2f kb = *(const v2f*)(krow + 4 * s + 2 * hi);
        acc = wmma4(qa[s], kb, acc);
      }
      const int msk = mask_s[k0 + m];        // col = k0+m for this lane
#pragma unroll
      for (int r = 0; r < 8; ++r) {          // C/D: row = r+8*hi, col = m
        float v = acc[r] * SCALE;
        if (msk == 0) v = NEGV;
        smem[(size_t)(r + 8 * hi) * S_ + k0 + m] = v;
      }
    }
  }
  __syncthreads();

  // ================= Phase 2: softmax + dropout, write p_attn ============
  {
    const int row = tid >> 4;                // 0..15, 16 lanes per row
    const int g   = tid & 15;
    float* srow = smem + (size_t)row * S_;

    float mx = -3.402823466e38f;
#pragma unroll 4
    for (int i = 0; i < 32; ++i) {
      const float4 v = *(const float4*)(srow + g * 4 + 64 * i);
      mx = fmaxf(mx, fmaxf(fmaxf(v.x, v.y), fmaxf(v.z, v.w)));
    }
#pragma unroll
    for (int off = 1; off < 16; off <<= 1)
      mx = fmaxf(mx, __shfl_xor(mx, off, 32));

    float ls = 0.0f;
#pragma unroll 4
    for (int i = 0; i < 32; ++i) {
      const float4 v = *(const float4*)(srow + g * 4 + 64 * i);
      ls += __expf(v.x - mx) + __expf(v.y - mx) +
            __expf(v.z - mx) + __expf(v.w - mx);
    }
#pragma unroll
    for (int off = 1; off < 16; off <<= 1)
      ls += __shfl_xor(ls, off, 32);
    const float inv = 1.0f / ls;

    const unsigned rbase =
        (unsigned)((bh * S_ + q0 + row)) * (unsigned)S_;  // < 2^27, fits u32
    float* prow = pattn + (size_t)row * S_;
#pragma unroll 2
    for (int i = 0; i < 32; ++i) {
      const int c = g * 4 + 64 * i;
      const float4 v = *(const float4*)(srow + c);
      float4 p;
      p.x = dropf(__expf(v.x - mx) * inv, rbase + c + 0);
      p.y = dropf(__expf(v.y - mx) * inv, rbase + c + 1);
      p.z = dropf(__expf(v.z - mx) * inv, rbase + c + 2);
      p.w = dropf(__expf(v.w - mx) * inv, rbase + c + 3);
      *(float4*)(srow + c) = p;              // for phase 3
      *(float4*)(prow + c) = p;              // p_attn output (single HBM pass)
    }
  }
  __syncthreads();

  // ================= Phase 3: out = P @ V ================================
  {
    const int nt = w & 3;                    // n-tile: cols n0..n0+15 of D
    const int n0 = nt * 16;
    const int kh = w >> 2;                   // k-half: 0 or 1
    v8f acc = {};
    const int s0 = kh * 256;
    for (int s = s0; s < s0 + 256; ++s) {    // k = 4s..4s+3
      const v2f a = *(const v2f*)(smem + (size_t)m * S_ + 4 * s + 2 * hi);
      v2f bv;
      bv.x = V[(size_t)(4 * s + 2 * hi) * D_ + n0 + m];
      bv.y = V[(size_t)(4 * s + 2 * hi + 1) * D_ + n0 + m];
      acc = wmma4(a, bv, acc);
    }
    if (kh == 1) {
#pragma unroll
      for (int r = 0; r < 8; ++r)
        red[nt * 256 + r * 32 + lane] = acc[r];
    }
    __syncthreads();
    if (kh == 0) {
#pragma unroll
      for (int r = 0; r < 8; ++r) {
        const float v = acc[r] + red[nt * 256 + r * 32 + lane];
        outp[(size_t)(r + 8 * hi) * D_ + n0 + m] = v;
      }
    }
  }
}

extern "C" void kernel_launch(void* const* d_in, const int* in_sizes, int n_in,
                              void* d_out, int out_size, void* d_ws, size_t ws_size,
                              hipStream_t stream) {
  (void)in_sizes; (void)n_in; (void)out_size; (void)d_ws; (void)ws_size;
  const float* Q = (const float*)d_in[0];
  const float* K = (const float*)d_in[1];
  const float* V = (const float*)d_in[2];
  const int*   M = (const int*)d_in[3];

  float* outp  = (float*)d_out;                               // [B,H,S,D]
  float* pattn = (float*)d_out + (size_t)B_ * H_ * S_ * D_;   // [B,H,S,S]

  const int blocks = B_ * H_ * (S_ / QT);                     // 4096
  const size_t smem_bytes = (size_t)(QT * S_ + S_) * sizeof(float); // 136 KB
  sdpa_f32_wmma_kernel<<<blocks, 256, smem_bytes, stream>>>(Q, K, V, M,
                                                            outp, pattn);
}